// DQN_1005022347920
// MI455X (gfx1250) — compile-verified
//
#include <hip/hip_runtime.h>
#include <hip/hip_bf16.h>
#include <limits.h>

typedef __attribute__((ext_vector_type(2))) float v2f;
typedef __attribute__((ext_vector_type(8))) float v8f;

#define NEG_SLOPE 0.2f

// Monotone float <-> signed-int key for atomicMax-based float max.
__device__ __forceinline__ int f2ord(float f) {
  int b = __float_as_int(f);
  return (b >= 0) ? b : (b ^ 0x7fffffff);
}
__device__ __forceinline__ float ord2f(int k) {
  int b = (k >= 0) ? k : (k ^ 0x7fffffff);
  return __int_as_float(b);
}

__global__ void fill_f32(float* __restrict__ p, float v, int n) {
  int i = blockIdx.x * blockDim.x + threadIdx.x;
  if (i < n) p[i] = v;
}
__global__ void fill_i32(int* __restrict__ p, int v, int n) {
  int i = blockIdx.x * blockDim.x + threadIdx.x;
  if (i < n) p[i] = v;
}

// Out[M,N] = (optional relu)(X[M,K] @ W[N,K]^T + bias[N]) using fp32 WMMA.
// One wave32 computes a 16x16 output tile; K stepped by 4 per V_WMMA_F32_16X16X4_F32.
// A layout: lane l -> M=l%16, regs hold K = r + (l<16?0:2)  (float2 load)
// B layout (mirrored): lane l -> N=l%16, regs hold K = r + (l<16?0:2)
// C/D layout: reg r, lane l -> M = r + (l<16?0:8), N = l%16
__global__ void gemm_wmma(const float* __restrict__ X, const float* __restrict__ W,
                          const float* __restrict__ bias, float* __restrict__ Out,
                          int M, int K, int N, int relu)
{
  int gid  = blockIdx.x * blockDim.x + threadIdx.x;
  int wave = gid >> 5;
  int lane = gid & 31;
  int ntile = N >> 4;
  int tm = wave / ntile;
  int tn = wave - tm * ntile;
  if (tm * 16 >= M) return;                 // wave-uniform: EXEC stays all-1s for WMMA
  int half = lane >> 4;                     // 0: lanes 0-15, 1: lanes 16-31
  int l16  = lane & 15;
  long rowA = (long)(tm * 16 + l16) * K;
  long rowB = (long)(tn * 16 + l16) * K;
  v8f acc = {};
  for (int k = 0; k < K; k += 4) {
    int kk = k + half * 2;
    v2f a = *(const v2f*)(X + rowA + kk);
    v2f b = *(const v2f*)(W + rowB + kk);
    acc = __builtin_amdgcn_wmma_f32_16x16x4_f32(false, a, false, b,
                                                (short)0, acc, false, false);
  }
  int n = tn * 16 + l16;
  float bv = bias[n];
#pragma unroll
  for (int r = 0; r < 8; ++r) {
    int m = tm * 16 + r + half * 8;
    float v = acc[r] + bv;
    if (relu) v = fmaxf(v, 0.f);
    Out[(long)m * N + n] = v;
  }
}

// Pass 1: per (edge,head) wave -> alpha = sum_c leakyrelu(xl[src]+xr[dst]+ea*We)*att
// and atomicMax of alpha into per-(dst,head) max (ordered-int key).
__global__ void edge_alpha_kernel(const float* __restrict__ xl, const float* __restrict__ xr,
                                  const float* __restrict__ ea, const float* __restrict__ We,
                                  const float* __restrict__ att,
                                  const int* __restrict__ src, const int* __restrict__ dst,
                                  float* __restrict__ alpha, int* __restrict__ amax,
                                  int E, int H, int C)
{
  int gid = blockIdx.x * blockDim.x + threadIdx.x;
  int w = gid >> 5, lane = gid & 31;
  if (w >= E * H) return;
  int e = w / H, h = w - e * H;
  int s = src[e], d = dst[e];
  int HC = H * C;
  float eav = ea[e];
  const float* pl = xl + (long)s * HC + h * C;
  const float* pr = xr + (long)d * HC + h * C;
  const float* pw = We + h * C;
  const float* pa = att + h * C;
  float sum = 0.f;
  for (int c = lane; c < C; c += 32) {
    float m = pl[c] + pr[c] + eav * pw[c];
    m = (m > 0.f) ? m : NEG_SLOPE * m;
    sum += m * pa[c];
  }
#pragma unroll
  for (int o = 16; o > 0; o >>= 1) sum += __shfl_xor(sum, o, 32);
  if (lane == 0) {
    alpha[(long)e * H + h] = sum;
    atomicMax(amax + (long)d * H + h, f2ord(sum));
  }
}

// Pass 2: ex = exp(alpha - max[dst]); accumulate denominator.
__global__ void edge_exp_kernel(float* __restrict__ alpha, const int* __restrict__ amax,
                                float* __restrict__ den, const int* __restrict__ dst,
                                int E, int H)
{
  int i = blockIdx.x * blockDim.x + threadIdx.x;
  if (i >= E * H) return;
  int e = i / H, h = i - e * H;
  int d = dst[e];
  float ex = expf(alpha[i] - ord2f(amax[(long)d * H + h]));
  alpha[i] = ex;
  atomicAdd(den + (long)d * H + h, ex);
}

// Pass 3: out[dst] += xl[src] * (ex / den[dst]); wave per (edge,head), lanes over C.
__global__ void edge_scatter_kernel(const float* __restrict__ xl, const float* __restrict__ alpha,
                                    const float* __restrict__ den,
                                    const int* __restrict__ src, const int* __restrict__ dst,
                                    float* __restrict__ out, int E, int H, int C)
{
  int gid = blockIdx.x * blockDim.x + threadIdx.x;
  int w = gid >> 5, lane = gid & 31;
  if (w >= E * H) return;
  int e = w / H, h = w - e * H;
  int s = src[e], d = dst[e];
  int HC = H * C;
  float a = alpha[(long)e * H + h] / den[(long)d * H + h];
  const float* pl = xl + (long)s * HC + h * C;
  float* po = out + (long)d * HC + h * C;
  for (int c = lane; c < C; c += 32)
    atomicAdd(po + c, pl[c] * a);
}

__global__ void bias_relu_kernel(float* __restrict__ h, const float* __restrict__ bias,
                                 int total, int HC)
{
  int i = blockIdx.x * blockDim.x + threadIdx.x;
  if (i >= total) return;
  float v = h[i] + bias[i % HC];
  h[i] = fmaxf(v, 0.f);
}

// gate[n] = dot(h[n], Wg) + bg; atomicMax into per-graph max. Wave per node.
__global__ void gate_kernel(const float* __restrict__ h, const float* __restrict__ Wg,
                            const float* __restrict__ bg, const int* __restrict__ batch,
                            float* __restrict__ gate, int* __restrict__ gmax, int N, int K)
{
  int gid = blockIdx.x * blockDim.x + threadIdx.x;
  int w = gid >> 5, lane = gid & 31;
  if (w >= N) return;
  const float* p = h + (long)w * K;
  float s = 0.f;
  for (int k = lane; k < K; k += 32) s += p[k] * Wg[k];
#pragma unroll
  for (int o = 16; o > 0; o >>= 1) s += __shfl_xor(s, o, 32);
  if (lane == 0) {
    s += bg[0];
    gate[w] = s;
    atomicMax(gmax + batch[w], f2ord(s));
  }
}

__global__ void pool_exp_kernel(float* __restrict__ gate, const int* __restrict__ gmax,
                                float* __restrict__ gden, const int* __restrict__ batch, int N)
{
  int i = blockIdx.x * blockDim.x + threadIdx.x;
  if (i >= N) return;
  int b = batch[i];
  float ex = expf(gate[i] - ord2f(gmax[b]));
  gate[i] = ex;
  atomicAdd(gden + b, ex);
}

__global__ void pool_sum_kernel(const float* __restrict__ h, const float* __restrict__ gate,
                                const float* __restrict__ gden, const int* __restrict__ batch,
                                float* __restrict__ pooled, int N, int K)
{
  int gid = blockIdx.x * blockDim.x + threadIdx.x;
  int w = gid >> 5, lane = gid & 31;
  if (w >= N) return;
  int b = batch[w];
  float wt = gate[w] / gden[b];
  const float* p = h + (long)w * K;
  float* po = pooled + (long)b * K;
  for (int k = lane; k < K; k += 32)
    atomicAdd(po + k, p[k] * wt);
}

// Tiny batched linear: Out[b*ostride + o] = relu?(In[b,:K] . W[o,:K] + bias[o])
__global__ void small_linear_kernel(const float* __restrict__ In, const float* __restrict__ W,
                                    const float* __restrict__ bias, float* __restrict__ Out,
                                    int Bn, int K, int O, int ostride, int relu)
{
  int i = blockIdx.x * blockDim.x + threadIdx.x;
  if (i >= Bn * O) return;
  int bb = i / O, o = i - bb * O;
  const float* p = In + (long)bb * K;
  const float* w = W + (long)o * K;
  float s = bias[o];
  for (int k = 0; k < K; ++k) s += p[k] * w[k];
  if (relu) s = fmaxf(s, 0.f);
  Out[(long)bb * ostride + o] = s;
}

static inline unsigned blk(long n, int t) { return (unsigned)((n + t - 1) / t); }

extern "C" void kernel_launch(void* const* d_in, const int* in_sizes, int n_in,
                              void* d_out, int out_size, void* d_ws, size_t ws_size,
                              hipStream_t stream) {
  (void)in_sizes; (void)n_in; (void)out_size; (void)ws_size;
  const int N = 20000, E = 320000, B = 32;

  const float* x     = (const float*)d_in[0];
  const int*   src   = (const int*)d_in[1];
  const int*   dst   = src + E;
  const float* ea    = (const float*)d_in[2];
  const float* agent = (const float*)d_in[3];
  const int*   batch = (const int*)d_in[4];
  // d_in[5] = num_graphs (compile-time constant B=32)
  auto P = [&](int i) { return (const float*)d_in[i]; };

  // ---- workspace carve ----
  float* ws = (float*)d_ws;
  size_t off = 0;
  auto take = [&](size_t n) { float* p = ws + off; off += n; return p; };
  float* xl     = take((size_t)N * 256);
  float* xr     = take((size_t)N * 256);
  float* h0     = take((size_t)N * 256);
  float* h1     = take((size_t)N * 256);
  float* alpha  = take((size_t)E * 4);
  float* den    = take((size_t)N * 4);
  int*   amax   = (int*)take((size_t)N * 4);
  float* gate   = take((size_t)N);
  float* gden   = take(B);
  int*   gmax   = (int*)take(B);
  float* pooled = take((size_t)B * 256);
  float* t0     = take((size_t)B * 256);
  float* t1     = take((size_t)B * 256);
  float* t2     = take((size_t)B * 256);
  float* fused  = take((size_t)B * 128);

  // ---- one GATv2 layer ----
  auto gat_layer = [&](const float* hin, int K, int H, int C,
                       const float* Wl, const float* bl, const float* Wr, const float* br,
                       const float* We, const float* att, const float* bias, float* out) {
    int HC = H * C;
    long gthr = (long)(N / 16) * (HC / 16) * 32;
    gemm_wmma<<<blk(gthr, 256), 256, 0, stream>>>(hin, Wl, bl, xl, N, K, HC, 0);
    gemm_wmma<<<blk(gthr, 256), 256, 0, stream>>>(hin, Wr, br, xr, N, K, HC, 0);
    fill_f32<<<blk((long)N * HC, 256), 256, 0, stream>>>(out, 0.f, N * HC);
    fill_f32<<<blk((long)N * H, 256), 256, 0, stream>>>(den, 0.f, N * H);
    fill_i32<<<blk((long)N * H, 256), 256, 0, stream>>>(amax, INT_MIN, N * H);
    long EH = (long)E * H;
    edge_alpha_kernel<<<blk(EH * 32, 256), 256, 0, stream>>>(xl, xr, ea, We, att, src, dst,
                                                             alpha, amax, E, H, C);
    edge_exp_kernel<<<blk(EH, 256), 256, 0, stream>>>(alpha, amax, den, dst, E, H);
    edge_scatter_kernel<<<blk(EH * 32, 256), 256, 0, stream>>>(xl, alpha, den, src, dst,
                                                               out, E, H, C);
    bias_relu_kernel<<<blk((long)N * HC, 256), 256, 0, stream>>>(out, bias, N * HC, HC);
  };

  // gat1: 4 -> 256 (H=4,C=64)   params 6..12
  gat_layer(x, 4, 4, 64, P(6), P(7), P(8), P(9), P(10), P(11), P(12), h0);
  // gat2: 256 -> 256 (H=4,C=64) params 13..19
  gat_layer(h0, 256, 4, 64, P(13), P(14), P(15), P(16), P(17), P(18), P(19), h1);
  // gat3: 256 -> 256 (H=2,C=128) params 20..26
  gat_layer(h1, 256, 2, 128, P(20), P(21), P(22), P(23), P(24), P(25), P(26), h0);

  // ---- attentional pooling (gate = P27,P28) ----
  fill_f32<<<1, 256, 0, stream>>>(gden, 0.f, B);
  fill_i32<<<1, 256, 0, stream>>>(gmax, INT_MIN, B);
  fill_f32<<<blk((long)B * 256, 256), 256, 0, stream>>>(pooled, 0.f, B * 256);
  gate_kernel<<<blk((long)N * 32, 256), 256, 0, stream>>>(h0, P(27), P(28), batch,
                                                          gate, gmax, N, 256);
  pool_exp_kernel<<<blk(N, 256), 256, 0, stream>>>(gate, gmax, gden, batch, N);
  pool_sum_kernel<<<blk((long)N * 32, 256), 256, 0, stream>>>(h0, gate, gden, batch,
                                                              pooled, N, 256);

  // ---- graph_fc (29..36): 256->256r ->256r ->128r ->64, into fused[:, :64] ----
  small_linear_kernel<<<blk(B * 256, 256), 256, 0, stream>>>(pooled, P(29), P(30), t0, B, 256, 256, 256, 1);
  small_linear_kernel<<<blk(B * 256, 256), 256, 0, stream>>>(t0, P(31), P(32), t1, B, 256, 256, 256, 1);
  small_linear_kernel<<<blk(B * 128, 256), 256, 0, stream>>>(t1, P(33), P(34), t0, B, 256, 128, 128, 1);
  small_linear_kernel<<<blk(B * 64, 256), 256, 0, stream>>>(t0, P(35), P(36), fused, B, 128, 64, 128, 0);

  // ---- agent_fc (37..44): into fused[:, 64:128] ----
  small_linear_kernel<<<blk(B * 256, 256), 256, 0, stream>>>(agent, P(37), P(38), t1, B, 256, 256, 256, 1);
  small_linear_kernel<<<blk(B * 256, 256), 256, 0, stream>>>(t1, P(39), P(40), t2, B, 256, 256, 256, 1);
  small_linear_kernel<<<blk(B * 128, 256), 256, 0, stream>>>(t2, P(41), P(42), t1, B, 256, 128, 128, 1);
  small_linear_kernel<<<blk(B * 64, 256), 256, 0, stream>>>(t1, P(43), P(44), fused + 64, B, 128, 64, 128, 0);

  // ---- fc_output (45..50): 128->256r ->128r ->10 ----
  small_linear_kernel<<<blk(B * 256, 256), 256, 0, stream>>>(fused, P(45), P(46), t0, B, 128, 256, 256, 1);
  small_linear_kernel<<<blk(B * 128, 256), 256, 0, stream>>>(t0, P(47), P(48), t1, B, 256, 128, 128, 1);
  small_linear_kernel<<<blk(B * 10, 256), 256, 0, stream>>>(t1, P(49), P(50), (float*)d_out, B, 128, 10, 10, 0);
}